// BiasAddLayerNormFuser_3693671875259
// MI455X (gfx1250) — compile-verified
//
#include <hip/hip_runtime.h>
#include <cstdint>

// Fused: t1 = x + bias + residual; LayerNorm(t1)*gamma+beta; fp8(e4m3) quantize;
// global amax of per-row fp8-rounded abs-max.
// Shapes: N=8192 rows, H=4096 cols, all f32 in. Memory-bound (~0.5 GB traffic).

#define NROWS 8192
#define HDIM  4096
#define TPB   256            // 8 wave32 per workgroup
#define EPT   16             // elements per thread: 256*16 = 4096

typedef float v4f __attribute__((ext_vector_type(4)));
typedef int   v4i __attribute__((ext_vector_type(4)));

#if __has_builtin(__builtin_amdgcn_global_load_async_to_lds_b128) && \
    __has_builtin(__builtin_amdgcn_s_wait_asynccnt)
#define USE_ASYNC_LDS 1
#else
#define USE_ASYNC_LDS 0
#endif

// ---- fp8 e4m3fn round-trip (value -> fp8 -> f32) -------------------------
__device__ __forceinline__ float quant_e4m3(float v) {
#if __has_builtin(__builtin_amdgcn_cvt_pk_fp8_f32) && \
    __has_builtin(__builtin_amdgcn_cvt_f32_fp8)
    int p = __builtin_amdgcn_cvt_pk_fp8_f32(v, v, 0, false);
    return __builtin_amdgcn_cvt_f32_fp8(p, 0);
#else
    // Software RNE fallback: 4-bit exp (bias 7), 3-bit mantissa, saturate 448.
    unsigned s = __float_as_uint(v) & 0x80000000u;
    float a = fabsf(v);
    a = fminf(a, 448.0f);
    if (a < 0.015625f) {                       // below min normal 2^-6
        a = rintf(a * 512.0f) * 0.001953125f;  // subnormal grid 2^-9
    } else {
        int e;
        (void)frexpf(a, &e);                   // a = m * 2^e, m in [0.5,1)
        a = ldexpf(rintf(ldexpf(a, 4 - e)), e - 4);  // 3-bit mantissa RNE
    }
    return __uint_as_float(s | __float_as_uint(a));
#endif
}

__global__ __launch_bounds__(TPB)
void fused_bias_ln_fp8_kernel(const float* __restrict__ x,
                              const float* __restrict__ bias,
                              const float* __restrict__ residual,
                              const float* __restrict__ gamma,
                              const float* __restrict__ beta,
                              const float* __restrict__ scale,
                              float* __restrict__ t1_out,
                              float* __restrict__ ln_out,
                              float* __restrict__ amax_out)
{
    const int  row  = blockIdx.x;
    const int  tid  = threadIdx.x;
    const int  col0 = tid * EPT;
    const size_t base = (size_t)row * HDIM + col0;

    __shared__ float wsum[8], wsq[8], wmax[8];

#if USE_ASYNC_LDS
    __shared__ v4f lx[HDIM / 4];
    __shared__ v4f lr[HDIM / 4];
    {
        const float* gx = x + base;
        const float* gr = residual + base;
#pragma unroll
        for (int k = 0; k < 4; ++k) {
            __builtin_amdgcn_global_load_async_to_lds_b128(
                (v4i*)(gx + 4 * k), (v4i*)(&lx[tid * 4 + k]), 0, 0);
            __builtin_amdgcn_global_load_async_to_lds_b128(
                (v4i*)(gr + 4 * k), (v4i*)(&lr[tid * 4 + k]), 0, 0);
        }
        __builtin_amdgcn_s_wait_asynccnt(0);
    }
    __syncthreads();
#endif

    // ---- Phase A: t1 = x + bias + residual; accumulate sum / sumsq -------
    v4f tv[4];
    float sum = 0.0f, sq = 0.0f;
#pragma unroll
    for (int k = 0; k < 4; ++k) {
#if USE_ASYNC_LDS
        v4f xv = lx[tid * 4 + k];
        v4f rv = lr[tid * 4 + k];
#else
        v4f xv = *(const v4f*)(x + base + 4 * k);
        v4f rv = *(const v4f*)(residual + base + 4 * k);
#endif
        v4f bv = *(const v4f*)(bias + col0 + 4 * k);
        v4f t  = xv + bv + rv;
        tv[k] = t;
        sum += t.x + t.y + t.z + t.w;
        sq   = fmaf(t.x, t.x, sq);
        sq   = fmaf(t.y, t.y, sq);
        sq   = fmaf(t.z, t.z, sq);
        sq   = fmaf(t.w, t.w, sq);
        // streaming output: non-temporal so we don't thrash L2
        __builtin_nontemporal_store(t, (v4f*)(t1_out + base + 4 * k));
    }

    // ---- block reduction (wave32 shuffle, then 8-entry LDS combine) ------
#pragma unroll
    for (int m = 16; m > 0; m >>= 1) {
        sum += __shfl_xor(sum, m, 32);
        sq  += __shfl_xor(sq,  m, 32);
    }
    const int wave = tid >> 5;
    const int lane = tid & 31;
    if (lane == 0) { wsum[wave] = sum; wsq[wave] = sq; }
    __syncthreads();
    float tot = 0.0f, tot2 = 0.0f;
#pragma unroll
    for (int i = 0; i < 8; ++i) { tot += wsum[i]; tot2 += wsq[i]; }

    const float inv_h = 1.0f / (float)HDIM;
    const float mean  = tot * inv_h;
    const float var   = fmaf(tot2, inv_h, -mean * mean);
    const float rstd  = rsqrtf(var + 1e-5f);
    const float sc    = scale[0];

    // ---- Phase B: normalize, affine, fp8 quantize, abs-max ---------------
    float amax = 0.0f;
#pragma unroll
    for (int k = 0; k < 4; ++k) {
        v4f gv = *(const v4f*)(gamma + col0 + 4 * k);
        v4f bb = *(const v4f*)(beta  + col0 + 4 * k);
        v4f t  = tv[k];
        v4f l;
        l.x = fmaf((t.x - mean) * rstd, gv.x, bb.x);
        l.y = fmaf((t.y - mean) * rstd, gv.y, bb.y);
        l.z = fmaf((t.z - mean) * rstd, gv.z, bb.z);
        l.w = fmaf((t.w - mean) * rstd, gv.w, bb.w);
        amax = fmaxf(amax, fmaxf(fmaxf(fabsf(l.x), fabsf(l.y)),
                                 fmaxf(fabsf(l.z), fabsf(l.w))));
        v4f q;
        q.x = quant_e4m3(l.x * sc);
        q.y = quant_e4m3(l.y * sc);
        q.z = quant_e4m3(l.z * sc);
        q.w = quant_e4m3(l.w * sc);
        __builtin_nontemporal_store(q, (v4f*)(ln_out + base + 4 * k));
    }

#pragma unroll
    for (int m = 16; m > 0; m >>= 1)
        amax = fmaxf(amax, __shfl_xor(amax, m, 32));
    if (lane == 0) wmax[wave] = amax;
    __syncthreads();
    if (tid == 0) {
        float rm = wmax[0];
#pragma unroll
        for (int i = 1; i < 8; ++i) rm = fmaxf(rm, wmax[i]);
        rm = quant_e4m3(rm);                     // reference round-trips via fp8
        // rm >= 0, so integer compare == float compare
        atomicMax((int*)amax_out, __float_as_int(rm));
    }
}

extern "C" void kernel_launch(void* const* d_in, const int* in_sizes, int n_in,
                              void* d_out, int out_size, void* d_ws, size_t ws_size,
                              hipStream_t stream) {
    const float* x        = (const float*)d_in[0];
    const float* bias     = (const float*)d_in[1];
    const float* residual = (const float*)d_in[2];
    const float* gamma    = (const float*)d_in[3];
    const float* beta     = (const float*)d_in[4];
    const float* scale    = (const float*)d_in[5];
    // d_in[6] (amax_tensor) is the aliased output; we write amax into d_out.

    float* out      = (float*)d_out;
    float* t1_out   = out;                                   // N*H f32
    float* ln_out   = out + (size_t)NROWS * HDIM;            // N*H (fp8 values as f32)
    float* amax_out = out + 2ull * NROWS * HDIM;             // scalar

    (void)hipMemsetAsync(amax_out, 0, sizeof(float), stream); // capture-safe init
    fused_bias_ln_fp8_kernel<<<NROWS, TPB, 0, stream>>>(
        x, bias, residual, gamma, beta, scale, t1_out, ln_out, amax_out);
}